// psm_query_54185307406442
// MI455X (gfx1250) — compile-verified
//
#include <hip/hip_runtime.h>
#include <hip/hip_bf16.h>

typedef __attribute__((ext_vector_type(2))) float v2f;
typedef __attribute__((ext_vector_type(8))) float v8f;

#define S_TOT 4096
#define KSEL  2048

// log2(e) / sqrt(A), A=2: folds both the attention scale and the
// natural-exp -> exp2 conversion into the WMMA B operand.
#define SCALE_LOG2 1.02017068616996945f

// ---------------------------------------------------------------------------
// Kernel 1: fused rank-2 flash-attention filter + exact top-k threshold.
// One workgroup (1024 threads = 32 waves) per (batch, cav) pair. 8 pairs.
// Each wave owns 128 rows, processed as 2 passes x 4 resident row-blocks
// (4 blocks of online-softmax state fits the VGPR budget at 8 waves/SIMD
// without spilling); column tiles stream and feed 4 back-to-back WMMAs.
// ---------------------------------------------------------------------------
__global__ __launch_bounds__(1024) void attn_filter_kernel(
    const float* __restrict__ psm, float* __restrict__ fm)
{
    const int pair = blockIdx.x;          // 0..7
    const int b  = pair >> 2;
    const int lc = pair & 3;              // cav index - 1
    const float* ego0 = psm + (size_t)((b * 5 + 0) * 2 + 0) * S_TOT;
    const float* ego1 = psm + (size_t)((b * 5 + 0) * 2 + 1) * S_TOT;
    const float* cav0 = psm + (size_t)((b * 5 + 1 + lc) * 2 + 0) * S_TOT;
    const float* cav1 = psm + (size_t)((b * 5 + 1 + lc) * 2 + 1) * S_TOT;

    __shared__ unsigned keys[S_TOT];      // 16 KB of 320 KB LDS
    __shared__ unsigned sh_prefix;
    __shared__ int sh_cnt;
    __shared__ int sh_k;

    const int tid  = threadIdx.x;
    const int wave = tid >> 5;
    const int lane = tid & 31;
    const bool lo  = (lane < 16);
    const int half = lo ? 0 : 8;          // which 8 tile columns this lane owns

    // ---- Phase A: per-row online softmax (log2 domain) ---------------------
    for (int g = 0; g < 2; ++g) {
        float mrun[4], srun[4], acc0[4], acc1[4];
        v2f Bv[4];
        #pragma unroll
        for (int t = 0; t < 4; ++t) {
            const int i = wave * 128 + (g * 4 + t) * 16 + (lane & 15);
            // B operand (4x16): K rows 0,1 = scaled ego features; K=2,3 pad 0
            Bv[t].x = lo ? ego0[i] * SCALE_LOG2 : 0.0f;
            Bv[t].y = lo ? ego1[i] * SCALE_LOG2 : 0.0f;
            mrun[t] = -3.0e38f; srun[t] = 0.0f; acc0[t] = 0.0f; acc1[t] = 0.0f;
        }

        #pragma unroll 1
        for (int jt = 0; jt < 256; ++jt) {
            const int jbase = jt * 16;

            // A operand (16x4): M rows = cav features of 16 cols; K=2,3 pad 0
            v2f Av;
            Av.x = lo ? cav0[jbase + lane] : 0.0f;
            Av.y = lo ? cav1[jbase + lane] : 0.0f;

            // cav values for this lane's 8 columns (broadcast float4 loads)
            const float4 v0a = *(const float4*)(cav0 + jbase + half);
            const float4 v0b = *(const float4*)(cav0 + jbase + half + 4);
            const float4 v1a = *(const float4*)(cav1 + jbase + half);
            const float4 v1b = *(const float4*)(cav1 + jbase + half + 4);

            #pragma unroll
            for (int t = 0; t < 4; ++t) {
                v8f c = {};
                // sim^T tile (log2 domain): lane L (<16) -> row, cols 0..7;
                //                           lane L+16    -> row, cols 8..15
                v8f d = __builtin_amdgcn_wmma_f32_16x16x4_f32(
                    false, Av, false, Bv[t], (short)0, c, false, false);

                float tmax = fmaxf(fmaxf(fmaxf(d[0], d[1]), fmaxf(d[2], d[3])),
                                   fmaxf(fmaxf(d[4], d[5]), fmaxf(d[6], d[7])));
                float nm   = fmaxf(mrun[t], tmax);
                float corr = __builtin_amdgcn_exp2f(mrun[t] - nm);
                srun[t] *= corr; acc0[t] *= corr; acc1[t] *= corr; mrun[t] = nm;

                float w;
                w = __builtin_amdgcn_exp2f(d[0] - nm); srun[t] += w; acc0[t] += w * v0a.x; acc1[t] += w * v1a.x;
                w = __builtin_amdgcn_exp2f(d[1] - nm); srun[t] += w; acc0[t] += w * v0a.y; acc1[t] += w * v1a.y;
                w = __builtin_amdgcn_exp2f(d[2] - nm); srun[t] += w; acc0[t] += w * v0a.z; acc1[t] += w * v1a.z;
                w = __builtin_amdgcn_exp2f(d[3] - nm); srun[t] += w; acc0[t] += w * v0a.w; acc1[t] += w * v1a.w;
                w = __builtin_amdgcn_exp2f(d[4] - nm); srun[t] += w; acc0[t] += w * v0b.x; acc1[t] += w * v1b.x;
                w = __builtin_amdgcn_exp2f(d[5] - nm); srun[t] += w; acc0[t] += w * v0b.y; acc1[t] += w * v1b.y;
                w = __builtin_amdgcn_exp2f(d[6] - nm); srun[t] += w; acc0[t] += w * v0b.z; acc1[t] += w * v1b.z;
                w = __builtin_amdgcn_exp2f(d[7] - nm); srun[t] += w; acc0[t] += w * v0b.w; acc1[t] += w * v1b.w;
            }
        }

        // merge the two j-halves of each row (lane L <-> lane L+16)
        #pragma unroll
        for (int t = 0; t < 4; ++t) {
            float m2  = __shfl_xor(mrun[t], 16, 32);
            float s2  = __shfl_xor(srun[t], 16, 32);
            float a02 = __shfl_xor(acc0[t], 16, 32);
            float a12 = __shfl_xor(acc1[t], 16, 32);
            float M   = fmaxf(mrun[t], m2);
            float e1  = __builtin_amdgcn_exp2f(mrun[t] - M);
            float e2  = __builtin_amdgcn_exp2f(m2 - M);
            float den = srun[t] * e1 + s2 * e2;
            float o0  = (acc0[t] * e1 + a02 * e2) / den;
            float o1  = (acc1[t] * e1 + a12 * e2) / den;
            // sigmoid is monotone -> threshold pre-sigmoid, same selected set
            float z   = fmaxf(o0, o1);
            if (lo) {
                unsigned u = __float_as_uint(z);
                u = (u & 0x80000000u) ? ~u : (u | 0x80000000u);  // orderable key
                keys[wave * 128 + (g * 4 + t) * 16 + lane] = u;
            }
        }
    }
    __syncthreads();

    // ---- Phase B: exact k-th largest via MSB->LSB radix select -------------
    if (tid == 0) { sh_prefix = 0u; sh_k = KSEL; }
    __syncthreads();
    for (int bit = 31; bit >= 0; --bit) {
        if (tid == 0) sh_cnt = 0;
        __syncthreads();
        const unsigned pref   = sh_prefix;
        const unsigned himask = (bit < 31) ? (0xFFFFFFFFu << (bit + 1)) : 0u;
        const unsigned bmask  = 1u << bit;
        int local = 0;
        #pragma unroll
        for (int q = 0; q < 4; ++q) {
            unsigned kv = keys[tid * 4 + q];
            if (((kv & himask) == pref) && (kv & bmask)) local++;
        }
        if (local) atomicAdd(&sh_cnt, local);
        __syncthreads();
        if (tid == 0) {
            if (sh_cnt >= sh_k) sh_prefix = pref | bmask;
            else                sh_k -= sh_cnt;
        }
        __syncthreads();
    }
    const unsigned thr = sh_prefix;       // exact key of the k-th largest value

    float* fmOut = fm + (size_t)pair * S_TOT;
    #pragma unroll
    for (int q = 0; q < 4; ++q) {
        int idx = tid * 4 + q;
        fmOut[idx] = (keys[idx] >= thr) ? 1.0f : 0.0f;
    }
}

// ---------------------------------------------------------------------------
// Kernel 2: out[:, :1] = x[:, :1]; out[:,1:] = x[:,1:] * fm * keep  (float4)
// ---------------------------------------------------------------------------
__global__ __launch_bounds__(256) void apply_mask_kernel(
    const float* __restrict__ x, const float* __restrict__ fm,
    const int* __restrict__ mask, float* __restrict__ out, int n4)
{
    int f = blockIdx.x * blockDim.x + threadIdx.x;
    if (f >= n4) return;

    const float4 xv = ((const float4*)x)[f];
    const int chw4 = 64 * S_TOT / 4;      // 65536 float4 per (b,l)
    int bl  = f / chw4;
    int rem = f - bl * chw4;
    int l   = bl % 5;
    int bb  = bl / 5;

    float4 o;
    if (l == 0) {
        o = xv;
    } else {
        float keep = (mask[bb * 5 + l] != 0) ? 1.0f : 0.0f;
        int s = (rem * 4) & (S_TOT - 1);  // h*64+w offset, 16B aligned
        const float4 fv = *(const float4*)(fm + (size_t)(bb * 4 + (l - 1)) * S_TOT + s);
        o.x = xv.x * fv.x * keep;
        o.y = xv.y * fv.y * keep;
        o.z = xv.z * fv.z * keep;
        o.w = xv.w * fv.w * keep;
    }
    ((float4*)out)[f] = o;
}

// ---------------------------------------------------------------------------
extern "C" void kernel_launch(void* const* d_in, const int* in_sizes, int n_in,
                              void* d_out, int out_size, void* d_ws, size_t ws_size,
                              hipStream_t stream) {
    const float* x   = (const float*)d_in[0];   // (2,5,64,64,64) f32
    const float* psm = (const float*)d_in[1];   // (2,5,2,64,64)  f32
    const int*   msk = (const int*)d_in[2];     // (2,5)          i32
    float* out = (float*)d_out;
    float* fm  = (float*)d_ws;                  // 8*4096 floats of scratch

    attn_filter_kernel<<<8, 1024, 0, stream>>>(psm, fm);

    const int n4 = out_size / 4;                // 655360 float4
    apply_mask_kernel<<<(n4 + 255) / 256, 256, 0, stream>>>(x, fm, msk, out, n4);
}